// AggregationLayer_67379446940091
// MI455X (gfx1250) — compile-verified
//
#include <hip/hip_runtime.h>

// CDNA5 / gfx1250: wave32, WMMA F32 16x16x4 (A,B = 2 VGPRs; C/D = 8 VGPRs)
typedef float v2f __attribute__((ext_vector_type(2)));
typedef float v8f __attribute__((ext_vector_type(8)));

#define DIM           128   // feature dimension (reference D)
#define SEGS_PER_TILE 16    // one WMMA M-tile of output segments
#define COL_SPLIT     4     // waves per tile; each wave covers 128/4 = 32 cols (2 N-tiles)

__device__ __forceinline__ int lower_bound_i32(const int* __restrict__ a, int n, int key) {
  int lo = 0, hi = n;
  while (lo < hi) {
    int mid = (lo + hi) >> 1;
    if (a[mid] < key) lo = mid + 1; else hi = mid;
  }
  return lo;
}

// Process one 4-row chunk: one-hot A (segment membership) x B (gathered rows, 2 col tiles).
__device__ __forceinline__ void chunk4(const float* __restrict__ values,
                                       const int*   __restrict__ seg,
                                       const int*   __restrict__ gidx,
                                       int r, int end_s, int rclamp,
                                       int base, int c0, int l15, int ksel,
                                       v8f& acc0, v8f& acc1, float& cnt)
{
  const int rk  = r + ksel;                    // even -> 8B-aligned int2 loads
  const int rkc = rk < rclamp ? rk : rclamp;   // v_min; clamp only affects invalid lanes
  const int2 s2 = *(const int2*)(seg  + rkc);
  const int2 g2 = *(const int2*)(gidx + rkc);

  // One-hot A: A[m][k] = (seg(row k) == base+m), masked for rows past `end_s`.
  // NOTE: bitwise '&' (not '&&') -> both loads stay unconditional, no exec-masked
  // predication blocks in the hot loop; lowers to v_cmp + s_and + v_cndmask.
  const int m0 = (int)(s2.x == base + l15) & (int)(rk     < end_s);
  const int m1 = (int)(s2.y == base + l15) & (int)(rk + 1 < end_s);
  v2f a;
  a.x = m0 ? 1.0f : 0.0f;
  a.y = m1 ? 1.0f : 0.0f;
  cnt += a.x + a.y;

  // 32-bit element offsets -> saddr + 32b voffset (scale_offset) loads, no 64b VALU math
  const unsigned off0 = (unsigned)(g2.x * DIM + c0 + l15);
  const unsigned off1 = (unsigned)(g2.y * DIM + c0 + l15);
  v2f b0, b1;
  b0.x = values[off0];       b0.y = values[off1];        // N-tile 0: cols c0..c0+15
  b1.x = values[off0 + 16];  b1.y = values[off1 + 16];   // N-tile 1: cols c0+16..c0+31

  acc0 = __builtin_amdgcn_wmma_f32_16x16x4_f32(false, a, false, b0, (short)0,
                                               acc0, false, false);
  acc1 = __builtin_amdgcn_wmma_f32_16x16x4_f32(false, a, false, b1, (short)0,
                                               acc1, false, false);
}

__global__ __launch_bounds__(256)
void seg_mean_wmma_kernel(const float* __restrict__ values,
                          const int*   __restrict__ gidx,
                          const int*   __restrict__ seg,
                          float*       __restrict__ out,
                          int total_inputs, int num_seg)
{
  // Force wave id into an SGPR so the guard below is a scalar branch (EXEC untouched).
  const int wid  = __builtin_amdgcn_readfirstlane((blockIdx.x * blockDim.x + threadIdx.x) >> 5);
  const int lane = threadIdx.x & 31;
  const int tiles = (num_seg + SEGS_PER_TILE - 1) / SEGS_PER_TILE;
  if (wid >= tiles * COL_SPLIT) return;                 // scalar, wave-uniform exit

  const int tile = wid / COL_SPLIT;
  const int base = tile * SEGS_PER_TILE;                // first segment of this tile
  const int c0   = (wid % COL_SPLIT) * 32;              // first column of this wave's slice

  // Row range [start, end) owned by this 16-segment tile (segment_ids are sorted).
  // readfirstlane makes the trip count scalar: uniform loop, full EXEC at every WMMA.
  int start = lower_bound_i32(seg, total_inputs, base);
  int end_s = lower_bound_i32(seg, total_inputs, base + SEGS_PER_TILE);
  start = __builtin_amdgcn_readfirstlane(start) & ~3;   // earlier rows: seg < base -> A==0
  end_s = __builtin_amdgcn_readfirstlane(end_s);

  const int l15    = lane & 15;        // M index (A) / N index (B,C,D) within tile
  const int ksel   = (lane >> 4) << 1; // lanes 0-15 hold K=0,1 ; lanes 16-31 hold K=2,3
  const int rclamp = (total_inputs - 2) & ~1;

  // Two independent accumulator sets -> two interleaved WMMA dependency chains.
  v8f accA0 = {0.f,0.f,0.f,0.f,0.f,0.f,0.f,0.f};
  v8f accA1 = {0.f,0.f,0.f,0.f,0.f,0.f,0.f,0.f};
  v8f accB0 = {0.f,0.f,0.f,0.f,0.f,0.f,0.f,0.f};
  v8f accB1 = {0.f,0.f,0.f,0.f,0.f,0.f,0.f,0.f};
  float cnt = 0.0f;

  for (int r = start; r < end_s; r += 8) {              // scalar loop, 8 rows / iter
    chunk4(values, seg, gidx, r,     end_s, rclamp, base, c0, l15, ksel, accA0, accA1, cnt);
    chunk4(values, seg, gidx, r + 4, end_s, rclamp, base, c0, l15, ksel, accB0, accB1, cnt);
  }
  const v8f acc0 = accA0 + accB0;
  const v8f acc1 = accA1 + accB1;

  // Per-segment counts: lane L(<16) summed K=0,1 indicators for segment L; lane L+16 K=2,3.
  const float cnt_tot = cnt + __shfl_xor(cnt, 16, 32);  // lane L -> count[L & 15]
  const float rcp = 1.0f / fmaxf(cnt_tot, 1.0f);        // segment-mean divisor (guarded)

  // C/D layout: VGPR j holds M = j (lanes 0-15) or M = j+8 (lanes 16-31), N = lane&15.
  #pragma unroll
  for (int j = 0; j < 8; ++j) {
    const int m = (lane < 16) ? j : (j + 8);
    const float d = __shfl(rcp, m, 32);                 // divisor for segment base+m
    const int row = base + m;
    if (row < num_seg) {
      // Output is written once and never re-read: non-temporal, keep L2 for `values`.
      __builtin_nontemporal_store(acc0[j] * d, out + (size_t)row * DIM + c0 +      l15);
      __builtin_nontemporal_store(acc1[j] * d, out + (size_t)row * DIM + c0 + 16 + l15);
    }
  }
}

extern "C" void kernel_launch(void* const* d_in, const int* in_sizes, int n_in,
                              void* d_out, int out_size, void* d_ws, size_t ws_size,
                              hipStream_t stream) {
  const float* values = (const float*)d_in[0];   // [N_SRC, 128] f32
  const int*   gidx   = (const int*)  d_in[1];   // [TOTAL_INPUTS] int
  const int*   seg    = (const int*)  d_in[2];   // [TOTAL_INPUTS] int, sorted
  float*       out    = (float*)      d_out;     // [num_segments, 128] f32

  const int total   = in_sizes[1];
  const int num_seg = out_size / DIM;            // avoids reading device scalar d_in[3]
  const int tiles   = (num_seg + SEGS_PER_TILE - 1) / SEGS_PER_TILE;
  const int waves   = tiles * COL_SPLIT;         // 10,000 waves for the reference shapes
  const int threads = 256;                       // 8 waves / block on wave32
  const int blocks  = (waves * 32 + threads - 1) / threads;

  seg_mean_wmma_kernel<<<blocks, threads, 0, stream>>>(values, gidx, seg, out,
                                                       total, num_seg);
}